// TransitionDown_7473243095309
// MI455X (gfx1250) — compile-verified
//
#include <hip/hip_runtime.h>

// Problem constants (match reference)
static constexpr int Bb    = 16;
static constexpr int C_IN  = 64;
static constexpr int C_OUT = 128;
static constexpr int Nn    = 8192;
static constexpr int Mm    = 2048;
static constexpr int Kk    = 16;

typedef float v2f __attribute__((ext_vector_type(2)));
typedef float v8f __attribute__((ext_vector_type(8)));

// ---------------------------------------------------------------------------
// Kernel 1: Farthest point sampling. One 1024-thread block per batch.
// Each thread keeps its 8 points (coords + running min-dist) in registers.
// Per step: update min-dist vs last pick, then argmax via wave shuffle +
// 32-partial LDS reduction. Tie-break prefers lower index (matches argmax).
// ---------------------------------------------------------------------------
__global__ __launch_bounds__(1024)
void fps_kernel(const float* __restrict__ coords, float* __restrict__ fps_coords)
{
    const int b   = blockIdx.x;
    const int tid = threadIdx.x;
    const float* cb = coords + (size_t)b * 3 * Nn;

    float px[8], py[8], pz[8], dmin[8];
#pragma unroll
    for (int i = 0; i < 8; ++i) {
        int n = tid + i * 1024;
        px[i] = cb[n];
        py[i] = cb[Nn + n];
        pz[i] = cb[2 * Nn + n];
        dmin[i] = 1e10f;
    }

    __shared__ float swv[32];
    __shared__ int   swi[32];
    __shared__ int   s_last;

    const int lane = tid & 31;
    const int wid  = tid >> 5;

    if (tid == 0) {  // m = 0 picks index 0
        fps_coords[((size_t)b * 3 + 0) * Mm] = cb[0];
        fps_coords[((size_t)b * 3 + 1) * Mm] = cb[Nn];
        fps_coords[((size_t)b * 3 + 2) * Mm] = cb[2 * Nn];
    }

    int last = 0;
    for (int m = 1; m < Mm; ++m) {
        const float lx = cb[last], ly = cb[Nn + last], lz = cb[2 * Nn + last];
        float bv = -1.0f; int bn = 0;
#pragma unroll
        for (int i = 0; i < 8; ++i) {
            float dx = px[i] - lx, dy = py[i] - ly, dz = pz[i] - lz;
            float d = dx * dx + dy * dy + dz * dz;
            dmin[i] = fminf(dmin[i], d);
            if (dmin[i] > bv) { bv = dmin[i]; bn = tid + i * 1024; }  // strict > keeps lower n
        }
        // wave32 argmax reduce, lower index wins ties
#pragma unroll
        for (int off = 16; off; off >>= 1) {
            float ov = __shfl_down(bv, off);
            int   on = __shfl_down(bn, off);
            if (ov > bv || (ov == bv && on < bn)) { bv = ov; bn = on; }
        }
        if (lane == 0) { swv[wid] = bv; swi[wid] = bn; }
        __syncthreads();
        if (wid == 0) {
            bv = swv[lane]; bn = swi[lane];
#pragma unroll
            for (int off = 16; off; off >>= 1) {
                float ov = __shfl_down(bv, off);
                int   on = __shfl_down(bn, off);
                if (ov > bv || (ov == bv && on < bn)) { bv = ov; bn = on; }
            }
            if (lane == 0) {
                s_last = bn;
                fps_coords[((size_t)b * 3 + 0) * Mm + m] = cb[bn];
                fps_coords[((size_t)b * 3 + 1) * Mm + m] = cb[Nn + bn];
                fps_coords[((size_t)b * 3 + 2) * Mm + m] = cb[2 * Nn + bn];
            }
        }
        __syncthreads();
        last = s_last;
    }
}

// ---------------------------------------------------------------------------
// Kernel 2: KNN. One thread per (b, m) center; candidates tiled through LDS.
// 16-entry insertion-sorted register list; strict '<' keeps lower-index ties.
// ---------------------------------------------------------------------------
__global__ __launch_bounds__(256)
void knn_kernel(const float* __restrict__ coords,
                const float* __restrict__ fps_coords,
                int* __restrict__ knn_idx)
{
    __shared__ float tsx[256], tsy[256], tsz[256];
    const int tid = threadIdx.x;
    const int gid = blockIdx.x * 256 + tid;        // [0, B*M)
    const int b = gid / Mm, m = gid % Mm;          // b uniform per block (2048 % 256 == 0)
    const float* cb = coords + (size_t)b * 3 * Nn;

    const float cx = fps_coords[((size_t)b * 3 + 0) * Mm + m];
    const float cy = fps_coords[((size_t)b * 3 + 1) * Mm + m];
    const float cz = fps_coords[((size_t)b * 3 + 2) * Mm + m];

    float bd[16]; int bi[16];
#pragma unroll
    for (int i = 0; i < 16; ++i) { bd[i] = 3.0e38f; bi[i] = 0; }

    for (int base = 0; base < Nn; base += 256) {
        tsx[tid] = cb[base + tid];
        tsy[tid] = cb[Nn + base + tid];
        tsz[tid] = cb[2 * Nn + base + tid];
        __syncthreads();
        for (int q = 0; q < 256; ++q) {
            float dx = cx - tsx[q], dy = cy - tsy[q], dz = cz - tsz[q];
            float d = dx * dx + dy * dy + dz * dz;
            if (d < bd[15]) {
                float vd = d; int vi = base + q;
#pragma unroll
                for (int jj = 0; jj < 16; ++jj) {
                    if (vd < bd[jj]) {
                        float td = bd[jj]; int ti = bi[jj];
                        bd[jj] = vd; bi[jj] = vi; vd = td; vi = ti;
                    }
                }
            }
        }
        __syncthreads();
    }
    int* out = knn_idx + (size_t)gid * Kk;
#pragma unroll
    for (int i = 0; i < 16; ++i) out[i] = bi[i];
}

// ---------------------------------------------------------------------------
// Kernel 3: fused gather + GEMM (v_wmma_f32_16x16x4_f32) + BN + ReLU + max-K.
// One wave per (b, m): H[128x16] = W[128x64] * X[64x16] via 8 row tiles x
// 16 K-chunks = 128 WMMA ops with 8 independent accumulators interleaved.
// W + folded BN affine staged in LDS once per 8-wave block.
// ---------------------------------------------------------------------------
__global__ __launch_bounds__(256)
void gemm_kernel(const float* __restrict__ x,
                 const int* __restrict__ knn_idx,
                 const float* __restrict__ conv_w,
                 const float* __restrict__ conv_b,
                 const float* __restrict__ bn_gamma,
                 const float* __restrict__ bn_beta,
                 const float* __restrict__ bn_mean,
                 const float* __restrict__ bn_var,
                 float* __restrict__ y)
{
    __shared__ __align__(16) float sW[C_OUT * C_IN];   // 32 KB
    __shared__ float sScale[C_OUT], sShift[C_OUT];

    const int tid = threadIdx.x;
    for (int i = tid; i < C_OUT * C_IN; i += 256) sW[i] = conv_w[i];
    if (tid < C_OUT) {
        float s = bn_gamma[tid] * rsqrtf(bn_var[tid] + 1e-5f);
        sScale[tid] = s;
        sShift[tid] = bn_beta[tid] - bn_mean[tid] * s + conv_b[tid] * s;
    }
    __syncthreads();

    const int w    = blockIdx.x * 8 + (tid >> 5);   // global wave id = b*M + m
    const int lane = tid & 31;
    const int b    = w / Mm;
    const int m    = w % Mm;
    const int hf   = lane >> 4;                     // half: lanes 16-31 -> 1
    const int j    = lane & 15;                     // column (neighbor) index / A row

    // Gather B panel: lane holds X[c, j] for c = 4t+2*hf, 4t+2*hf+1 (t=0..15)
    const int gidx = knn_idx[((size_t)b * Mm + m) * Kk + j];
    const float* xb = x + (size_t)b * C_IN * Nn + gidx;
    v2f bfrag[16];
#pragma unroll
    for (int t = 0; t < 16; ++t) {
        int c0 = 4 * t + 2 * hf;
        bfrag[t][0] = xb[(size_t)c0 * Nn];
        bfrag[t][1] = xb[(size_t)(c0 + 1) * Nn];
    }

    v8f acc[8];
    const v8f zero = {};
#pragma unroll
    for (int tl = 0; tl < 8; ++tl) acc[tl] = zero;

#pragma unroll
    for (int t = 0; t < 16; ++t) {
        const v2f bf = bfrag[t];
#pragma unroll
        for (int tl = 0; tl < 8; ++tl) {
            // A frag: lane holds W[tl*16 + j, 4t+2*hf .. +1] (8B-aligned LDS pair)
            const v2f a = *(const v2f*)&sW[(tl * 16 + j) * C_IN + 4 * t + 2 * hf];
            acc[tl] = __builtin_amdgcn_wmma_f32_16x16x4_f32(
                false, a, false, bf, (short)0, acc[tl], false, false);
        }
    }

    // Epilogue: affine(BN+bias) -> ReLU -> max over the 16 neighbors (columns),
    // i.e. reduce across each 16-lane half with xor masks 1,2,4,8.
    float* yb = y + (size_t)b * C_OUT * Mm + m;
#pragma unroll
    for (int tl = 0; tl < 8; ++tl) {
#pragma unroll
        for (int i = 0; i < 8; ++i) {
            int o = tl * 16 + i + 8 * hf;
            float v = acc[tl][i] * sScale[o] + sShift[o];
            v = fmaxf(v, 0.0f);
#pragma unroll
            for (int off = 1; off < 16; off <<= 1)
                v = fmaxf(v, __shfl_xor(v, off, 32));
            if (j == 0) yb[(size_t)o * Mm] = v;
        }
    }
}

// ---------------------------------------------------------------------------
extern "C" void kernel_launch(void* const* d_in, const int* in_sizes, int n_in,
                              void* d_out, int out_size, void* d_ws, size_t ws_size,
                              hipStream_t stream)
{
    const float* x        = (const float*)d_in[0];
    const float* coords   = (const float*)d_in[1];
    const float* conv_w   = (const float*)d_in[2];
    const float* conv_b   = (const float*)d_in[3];
    const float* bn_gamma = (const float*)d_in[4];
    const float* bn_beta  = (const float*)d_in[5];
    const float* bn_mean  = (const float*)d_in[6];
    const float* bn_var   = (const float*)d_in[7];

    float* y          = (float*)d_out;                       // [B, C_OUT, M]
    float* fps_coords = y + (size_t)Bb * C_OUT * Mm;         // [B, 3, M]
    int*   knn_idx    = (int*)d_ws;                          // [B, M, K] (2 MB)

    fps_kernel<<<Bb, 1024, 0, stream>>>(coords, fps_coords);
    knn_kernel<<<(Bb * Mm) / 256, 256, 0, stream>>>(coords, fps_coords, knn_idx);
    gemm_kernel<<<(Bb * Mm) / 8, 256, 0, stream>>>(x, knn_idx, conv_w, conv_b,
                                                   bn_gamma, bn_beta, bn_mean, bn_var, y);
}